// AudioVideoInter_85667417686179
// MI455X (gfx1250) — compile-verified
//
#include <hip/hip_runtime.h>

// Problem shape (fixed by the reference setup).
constexpr int T = 1024;
constexpr int B = 32;
constexpr int C = 512;

typedef __attribute__((ext_vector_type(2))) float v2f;
typedef __attribute__((ext_vector_type(8))) float v8f;

// ---------------------------------------------------------------------------
// Phase 1: means[r] = (1/C) * sum_c video[r*C + c], r = j*B + b (32768 rows).
// Row-sum via V_WMMA_F32_16X16X4_F32 with an all-ones B matrix:
//   D[m,n] = sum_k A[m,k] * 1  -> every column of D holds the row sum.
// A-matrix 16x4 f32 layout: lanes 0-15 hold (K=0,K=1), lanes 16-31 hold
// (K=2,K=3) for rows M = lane&15, so each lane streams v2f loads down one row.
// 16 rows per wave, 128 WMMA accumulations cover C=512.
// ---------------------------------------------------------------------------
__global__ void __launch_bounds__(256)
means_wmma_kernel(const float* __restrict__ video, float* __restrict__ means) {
  const int lane = threadIdx.x & 31;
  const int wave = (blockIdx.x * blockDim.x + threadIdx.x) >> 5;
  const int base = wave * 16;                 // first of 16 rows for this wave
  const int row  = base + (lane & 15);
  const int koff = (lane >> 4) * 2;           // lanes 16-31 take K=2,3

  const float* src = video + (size_t)row * C + koff;

  v2f ones; ones[0] = 1.0f; ones[1] = 1.0f;
  v8f acc = {};

#pragma unroll 4
  for (int k = 0; k < C; k += 4) {
    v2f a = *(const v2f*)(src + k);
    // 8 args: (neg_a, A, neg_b, B, c_mod, C, reuse_a, reuse_b)
    acc = __builtin_amdgcn_wmma_f32_16x16x4_f32(
        false, a, false, ones, (short)0, acc, false, false);
  }

  // D layout: lane 0 VGPR i = row-sum of row (base+i); lane 16 VGPR i = base+8+i.
  const float inv = 1.0f / (float)C;
  if (lane == 0) {
#pragma unroll
    for (int i = 0; i < 8; ++i) means[base + i] = acc[i] * inv;
  } else if (lane == 16) {
#pragma unroll
    for (int i = 0; i < 8; ++i) means[base + 8 + i] = acc[i] * inv;
  }
}

// ---------------------------------------------------------------------------
// Phase 2: one block per batch b.
//   vm[m] = means at the m-th j with labels[b,j]==1 (ascending j), t = count.
//   scale[p] = prod_{m=lo..hi} vm[m], lo = max(0, p-T+t), hi = min(p, t-1).
// Stored p-major: scale_pb[p*B + b].
// ---------------------------------------------------------------------------
__global__ void __launch_bounds__(T)
scan_scale_kernel(const float* __restrict__ means_jb,
                  const int* __restrict__ labels,
                  float* __restrict__ scale_pb) {
  const int b = blockIdx.x;
  const int p = threadIdx.x;

  __shared__ float vm[T];
  __shared__ unsigned char lab[T];
  __shared__ int t_sh;

  lab[p] = (labels[b * T + p] == 1) ? 1 : 0;
  __syncthreads();

  if (p == 0) {
    int cnt = 0;
    for (int j = 0; j < T; ++j) {
      if (lab[j]) vm[cnt++] = means_jb[j * B + b];
    }
    t_sh = cnt;
  }
  __syncthreads();

  const int t = t_sh;
  int lo = p - (T - t);
  if (lo < 0) lo = 0;
  int hi = (p < t - 1) ? p : (t - 1);

  float s = 1.0f;
  for (int m = lo; m <= hi; ++m) s *= vm[m];  // ascending m, matches jnp.prod
  scale_pb[p * B + b] = s;
}

// ---------------------------------------------------------------------------
// Phase 3: out[p,b,c] = audio[p,b,c] * scale_pb[p*B + b], float4 streaming.
// C/4 = 128 -> scale index = vec_idx >> 7. Prefetch ahead (global_prefetch_b8).
// ---------------------------------------------------------------------------
__global__ void __launch_bounds__(256)
scale_mul_kernel(const float* __restrict__ audio,
                 const float* __restrict__ scale_pb,
                 float* __restrict__ out) {
  const size_t N4 = (size_t)T * B * C / 4;
  const size_t i4 = (size_t)blockIdx.x * blockDim.x + threadIdx.x;

  const float4* a4 = (const float4*)audio;
  float4* o4 = (float4*)out;

  // Prefetch ~32KB ahead into the cache hierarchy (emits global_prefetch_b8).
  const size_t pf = i4 + 2048;
  if (pf < N4) __builtin_prefetch((const void*)(a4 + pf), 0, 1);

  float4 a = a4[i4];
  const float s = scale_pb[i4 >> 7];  // (p*B + b)
  float4 o;
  o.x = a.x * s;
  o.y = a.y * s;
  o.z = a.z * s;
  o.w = a.w * s;
  o4[i4] = o;
}

extern "C" void kernel_launch(void* const* d_in, const int* in_sizes, int n_in,
                              void* d_out, int out_size, void* d_ws, size_t ws_size,
                              hipStream_t stream) {
  const float* video  = (const float*)d_in[0];  // (T,B,C) f32
  const float* audio  = (const float*)d_in[1];  // (T,B,C) f32
  const int*   labels = (const int*)d_in[2];    // (B,T) i32
  float* out = (float*)d_out;                   // (T,B,C) f32

  float* means = (float*)d_ws;       // T*B floats, indexed j*B+b
  float* scale = means + (size_t)T * B;  // T*B floats, indexed p*B+b

  // Phase 1: 32768 rows / 16 rows-per-wave = 2048 waves = 256 blocks x 8 waves.
  means_wmma_kernel<<<256, 256, 0, stream>>>(video, means);

  // Phase 2: one block per batch.
  scan_scale_kernel<<<B, T, 0, stream>>>(means, labels, scale);

  // Phase 3: N/4 float4 elements.
  const int n4 = T * B * C / 4;  // 4,194,304
  scale_mul_kernel<<<n4 / 256, 256, 0, stream>>>(audio, scale, out);
}